// DogDetector_3839700762850
// MI455X (gfx1250) — compile-verified
//
#include <hip/hip_runtime.h>
#include <stdint.h>

#define BATCH   32
#define NANCH   102400
#define KCAND   1024
#define TOPK    1000
#define NBINS   2048
#define CAP     2048
#define MAXDET  100
#define XFORM_CLIP 4.135f

typedef __attribute__((ext_vector_type(2))) float v2f;
typedef __attribute__((ext_vector_type(8))) float v8f;

__device__ __forceinline__ float sigmoidf_(float x) { return 1.0f / (1.0f + expf(-x)); }

// ---------------------------------------------------------------------------
// Kernel 1: per-image histogram select of top-1000 scores, bitonic sort in
// LDS, decode boxes ONLY for the survivors (avoids 52MB of decoded boxes).
// One 1024-thread block (32 waves) per image.
// ---------------------------------------------------------------------------
__global__ __launch_bounds__(1024) void topk_decode_kernel(
    const float* __restrict__ bbox_pred, const float* __restrict__ conf_pred,
    const float* __restrict__ anchors, float4* __restrict__ cand_boxes,
    float* __restrict__ cand_scores)
{
    __shared__ unsigned hist[NBINS];
    __shared__ float ss[CAP];
    __shared__ int   ii[CAP];
    __shared__ int      sh_thr;
    __shared__ unsigned sh_cnt;

    const int img = blockIdx.x;
    const int tid = threadIdx.x;
    const float* conf = conf_pred + (size_t)img * NANCH;

    for (int b = tid; b < NBINS; b += 1024) hist[b] = 0u;
    if (tid == 0) sh_cnt = 0u;
    __syncthreads();

    // Pass 1: histogram of sigmoid scores in (0.5, 1.0]
    for (int i = tid; i < NANCH; i += 1024) {
        float c = conf[i];
        if (c > 0.0f) {                      // sigmoid(c) > 0.5  <=>  c > 0
            float s = sigmoidf_(c);
            int b = (int)((s - 0.5f) * 4096.0f);
            b = b < 0 ? 0 : (b > NBINS - 1 ? NBINS - 1 : b);
            atomicAdd(&hist[b], 1u);
        }
    }
    __syncthreads();

    // Find threshold bin (suffix scan from top)
    if (tid == 0) {
        unsigned total = 0; int thr = 0;
        for (int b = NBINS - 1; b >= 0; --b) {
            total += hist[b];
            if (total >= TOPK) { thr = b; break; }
        }
        sh_thr = thr;
    }
    __syncthreads();
    const int thr = sh_thr;

    // Pass 2: compact candidates with bin >= thr
    for (int i = tid; i < NANCH; i += 1024) {
        float c = conf[i];
        if (c > 0.0f) {
            float s = sigmoidf_(c);
            int b = (int)((s - 0.5f) * 4096.0f);
            b = b < 0 ? 0 : (b > NBINS - 1 ? NBINS - 1 : b);
            if (b >= thr) {
                unsigned p = atomicAdd(&sh_cnt, 1u);
                if (p < CAP) { ss[p] = s; ii[p] = i; }
            }
        }
    }
    __syncthreads();
    unsigned cnt = sh_cnt; if (cnt > CAP) cnt = CAP;
    for (int p = tid; p < CAP; p += 1024)
        if ((unsigned)p >= cnt) { ss[p] = 0.0f; ii[p] = -1; }
    __syncthreads();

    // Bitonic sort, descending by score, ties broken by lower index first
    for (unsigned k = 2; k <= CAP; k <<= 1) {
        for (unsigned j = k >> 1; j > 0; j >>= 1) {
            for (unsigned t = tid; t < CAP; t += 1024) {
                unsigned x = t ^ j;
                if (x > t) {
                    float sa = ss[t], sb = ss[x];
                    int   ia = ii[t], ib = ii[x];
                    bool aAfterB = (sa < sb) || (sa == sb && ia > ib);
                    if (((t & k) == 0u) ? aAfterB : !aAfterB) {
                        ss[t] = sb; ss[x] = sa; ii[t] = ib; ii[x] = ia;
                    }
                }
            }
            __syncthreads();
        }
    }

    // Pass 3: decode only the top-1000 boxes; pad [1000,1024) with zeros
    {
        const int k = tid;
        float4 box = make_float4(0.f, 0.f, 0.f, 0.f);
        float sc = 0.0f;
        if (k < TOPK) {
            int id = ii[k];
            if (id >= 0) {
                sc = ss[k];
                float4 d = reinterpret_cast<const float4*>(bbox_pred)[(size_t)img * NANCH + id];
                float4 a = reinterpret_cast<const float4*>(anchors)[id];
                float w  = a.z - a.x, h = a.w - a.y;
                float cx = a.x + 0.5f * w, cy = a.y + 0.5f * h;
                float dw = fminf(fmaxf(d.z, -XFORM_CLIP), XFORM_CLIP);
                float dh = fminf(fmaxf(d.w, -XFORM_CLIP), XFORM_CLIP);
                float pcx = d.x * w + cx, pcy = d.y * h + cy;
                float pw  = expf(dw) * w, ph = expf(dh) * h;
                box = make_float4(pcx - 0.5f * pw, pcy - 0.5f * ph,
                                  pcx + 0.5f * pw, pcy + 0.5f * ph);
            }
        }
        cand_boxes [(size_t)img * KCAND + k] = box;
        cand_scores[(size_t)img * KCAND + k] = sc;
    }
}

// ---------------------------------------------------------------------------
// Kernel 2: suppression bit-matrix.  Each wave computes a 16-row x 32-col tile
// of (iou > 0.5 && j > i).  The denominator term areaA[i]+areaB[j] comes from
// the matrix unit (V_WMMA_F32_16X16X4_F32 rank-2 outer product), the column
// boxes are staged LDS-direct with GLOBAL_LOAD_ASYNC_TO_LDS_B128 (ASYNCcnt),
// and the threshold test is division-free: iou>0.5  <=>  3*inter > sum + eps.
// Ballot packs the WMMA C-layout straight into row-major 32-bit mask words.
// ---------------------------------------------------------------------------
__global__ __launch_bounds__(256) void iou_mask_kernel(
    const float4* __restrict__ cand_boxes, uint32_t* __restrict__ mask)
{
    __shared__ float4 Ab[16];  __shared__ float Aa[16];
    __shared__ float4 Bb[256]; __shared__ float Ba[256];

    const int gid = blockIdx.x;
    const int img = gid >> 8;          // 256 blocks per image
    const int rem = gid & 255;
    const int ti  = rem >> 2;          // row tile 0..63
    const int cg  = rem & 3;           // col group 0..3 (256 cols each)
    const int tid = threadIdx.x;
    const int r0  = ti * 16;
    const int cbase = cg * 256;

    // Async copy this block's 256 column boxes global -> LDS (no VGPR bounce).
    {
        uint64_t gaddr = (uint64_t)(uintptr_t)(cand_boxes + (size_t)img * KCAND + cbase + tid);
        uint32_t laddr = (uint32_t)(uintptr_t)(&Bb[tid]);   // LDS offset = addr[31:0]
        asm volatile("global_load_async_to_lds_b128 %0, %1, off"
                     :: "v"(laddr), "v"(gaddr) : "memory");
    }
    // Row boxes (16 per block) via a normal load.
    float4 arow;
    if (tid < 16) arow = cand_boxes[(size_t)img * KCAND + r0 + tid];
    asm volatile("s_wait_asynccnt 0" ::: "memory");
    // Areas from the freshly landed LDS data (own-lane element).
    {
        float4 b = Bb[tid];
        Ba[tid] = (b.z - b.x) * (b.w - b.y);
        if (tid < 16) {
            Ab[tid] = arow;
            Aa[tid] = (arow.z - arow.x) * (arow.w - arow.y);
        }
    }
    __syncthreads();

    const int  wv = tid >> 5;          // wave 0..7 -> col word within group
    const int  l  = tid & 31;          // lane
    const int  cw = cg * 8 + wv;       // global col-word 0..31
    const int  lc = l & 15;
    const bool lo = (l < 16);

    // A: lanes 0-15 hold (M=lane, K0,K1); lanes 16-31 hold K2,K3 (zero)
    v2f afrag; afrag[0] = lo ? Aa[l] : 0.0f;  afrag[1] = lo ? 1.0f : 0.0f;
    // B: lanes 0-15 hold (N=lane, K0,K1); lanes 16-31 hold K2,K3 (zero)
    v2f bf0;   bf0[0]   = lo ? 1.0f : 0.0f;   bf0[1]   = lo ? Ba[wv * 32 + lc]      : 0.0f;
    v2f bf1;   bf1[0]   = lo ? 1.0f : 0.0f;   bf1[1]   = lo ? Ba[wv * 32 + 16 + lc] : 0.0f;
    v8f cz = {};
    v8f d0 = __builtin_amdgcn_wmma_f32_16x16x4_f32(false, afrag, false, bf0, (short)0, cz, false, false);
    v8f d1 = __builtin_amdgcn_wmma_f32_16x16x4_f32(false, afrag, false, bf1, (short)0, cz, false, false);

    const int moff = lo ? 0 : 8;
    float4 bB0 = Bb[wv * 32 + lc];
    float4 bB1 = Bb[wv * 32 + 16 + lc];

    uint32_t myword = 0u;
    #pragma unroll
    for (int k = 0; k < 8; ++k) {
        int m = k + moff;
        float4 bA = Ab[m];
        float ix = fminf(bA.z, bB0.z) - fmaxf(bA.x, bB0.x);
        float iy = fminf(bA.w, bB0.w) - fmaxf(bA.y, bB0.y);
        float inter0 = fmaxf(ix, 0.0f) * fmaxf(iy, 0.0f);
        float jx = fminf(bA.z, bB1.z) - fmaxf(bA.x, bB1.x);
        float jy = fminf(bA.w, bB1.w) - fmaxf(bA.y, bB1.y);
        float inter1 = fmaxf(jx, 0.0f) * fmaxf(jy, 0.0f);
        // iou > 0.5  <=>  inter > 0.5*(sum - inter + eps)  <=>  3*inter > sum + eps
        bool p0 = 3.0f * inter0 > d0[k] + 1e-9f;
        bool p1 = 3.0f * inter1 > d1[k] + 1e-9f;
        uint32_t balA = (uint32_t)__ballot(p0);   // bits0-15: row k cols0-15; bits16-31: row k+8 cols0-15
        uint32_t balB = (uint32_t)__ballot(p1);   // same for cols 16-31
        uint32_t wlow  = (balA & 0xFFFFu) | ((balB & 0xFFFFu) << 16);  // row r0+k
        uint32_t whigh = (balA >> 16)     | (balB & 0xFFFF0000u);      // row r0+k+8
        if (l == k)     myword = wlow;
        if (l == k + 8) myword = whigh;
    }

    if (lo) {
        int r = r0 + l;
        int base_c = cw * 32;
        uint32_t vm;
        if (r < base_c) vm = 0xFFFFFFFFu;
        else { int dlt = r - base_c; vm = (dlt >= 31) ? 0u : (0xFFFFFFFFu << (dlt + 1)); }
        mask[(((size_t)img * KCAND) + r) * 32 + cw] = myword & vm;
    }
}

// ---------------------------------------------------------------------------
// Kernel 3: sequential NMS suppression scan (one wave per image, lane j owns
// suppression word j); mask rows prefetched 16-deep to hide L2 latency.
// Then wave-cooperative ballot/popcount compaction of the first 100 kept.
// ---------------------------------------------------------------------------
__global__ __launch_bounds__(32) void nms_scan_kernel(
    const uint32_t* __restrict__ mask, const float4* __restrict__ cand_boxes,
    const float* __restrict__ cand_scores, float* __restrict__ out)
{
    const int img = blockIdx.x;
    const int l   = threadIdx.x;
    const uint32_t* mrow = mask + (size_t)img * KCAND * 32;

    uint32_t remv = 0u;   // lane l holds suppression bits for candidates [32l, 32l+31]
    for (int ch = 0; ch < KCAND / 16; ++ch) {
        uint32_t m[16];
        #pragma unroll
        for (int t = 0; t < 16; ++t) m[t] = mrow[(size_t)(ch * 16 + t) * 32 + l];
        #pragma unroll
        for (int t = 0; t < 16; ++t) {
            int i = ch * 16 + t;
            uint32_t rv = (uint32_t)__shfl((int)remv, i >> 5, 32);
            if (((rv >> (i & 31)) & 1u) == 0u) remv |= m[t];   // i is kept -> it suppresses
        }
    }

    float* o = out + (size_t)img * MAXDET * 5;
    for (int x = l; x < MAXDET * 5; x += 32) o[x] = 0.0f;

    int cnt = 0;
    for (int base = 0; base < KCAND && cnt < MAXDET; base += 32) {
        int i = base + l;
        float s = cand_scores[(size_t)img * KCAND + i];
        uint32_t rv = (uint32_t)__shfl((int)remv, base >> 5, 32);
        bool pred = (((rv >> l) & 1u) == 0u) && (s > 0.0f);
        uint32_t bal = (uint32_t)__ballot(pred);
        int pos = cnt + __popc(bal & ((1u << l) - 1u));
        if (pred && pos < MAXDET) {
            float4 b = cand_boxes[(size_t)img * KCAND + i];
            o[pos * 5 + 0] = b.x; o[pos * 5 + 1] = b.y;
            o[pos * 5 + 2] = b.z; o[pos * 5 + 3] = b.w;
            o[pos * 5 + 4] = s;
        }
        cnt += __popc(bal);
    }
}

// ---------------------------------------------------------------------------
extern "C" void kernel_launch(void* const* d_in, const int* in_sizes, int n_in,
                              void* d_out, int out_size, void* d_ws, size_t ws_size,
                              hipStream_t stream)
{
    const float* bbox_pred = (const float*)d_in[0];   // [32,102400,4]
    const float* conf_pred = (const float*)d_in[1];   // [32,102400]
    const float* anchors   = (const float*)d_in[2];   // [102400,4]
    float* out = (float*)d_out;                       // [32,100,5]

    char* ws = (char*)d_ws;
    float4*   cand_boxes  = (float4*)ws;                                   // 32*1024*16  = 524288 B
    float*    cand_scores = (float*)(ws + 524288);                         // 32*1024*4   = 131072 B
    uint32_t* mask        = (uint32_t*)(ws + 524288 + 131072);             // 32*1024*32*4 = 4 MiB

    topk_decode_kernel<<<BATCH, 1024, 0, stream>>>(bbox_pred, conf_pred, anchors,
                                                   cand_boxes, cand_scores);
    iou_mask_kernel<<<BATCH * 256, 256, 0, stream>>>(cand_boxes, mask);
    nms_scan_kernel<<<BATCH, 32, 0, stream>>>(mask, cand_boxes, cand_scores, out);
}